// SubtitleColorConsistencyLoss_63471026700870
// MI455X (gfx1250) — compile-verified
//
#include <hip/hip_runtime.h>
#include <hip/hip_bf16.h>
#include <math.h>

// ---------------- problem constants (match reference) ----------------
#define NB 8
#define CCH 32
#define HH 512
#define WWD 512
#define KK 6
#define HWPX (HH * WWD)          // 262144 pixels per image
#define WPI 256                  // waves per image
#define CHUNK (HWPX / WPI)       // 1024 pixels per wave (128 unrolled iters)

#define EPSV 1e-6f
#define MINPIX 20.0f

typedef __attribute__((ext_vector_type(2))) float v2f;
typedef __attribute__((ext_vector_type(4))) float v4f;
typedef __attribute__((ext_vector_type(4))) int   v4i;
typedef __attribute__((ext_vector_type(8))) float v8f;

// ws float layout: S1 [N][16][32] | SQ [N][16][32] | CNT [N][16]
#define WS_S1  0
#define WS_SQ  (NB * 16 * 32)
#define WS_CNT (2 * NB * 16 * 32)
#define WS_TOT (2 * NB * 16 * 32 + NB * 16)

// ---------------- kernel 0: zero accumulators ----------------
__global__ void scc_zero(float* __restrict__ ws) {
    int i = blockIdx.x * blockDim.x + threadIdx.x;
    if (i < WS_TOT) ws[i] = 0.0f;
}

// ---------------- kernel 1: fused segment-reduce via WMMA ----------------
// One wave per (image, pixel-chunk). Iteration = 8 pixels = 2 WMMA steps.
//   A (16x4) = one-hot class matrix: rows 0..5 = component (label==row+1),
//              row 6 = sub, row 7 = bg(!sub&&!text), row 8 = scene(text&&!sub)
//   B (4x16) = color[channel tile][pixel]; two tiles cover C=32.
//   D += A*B accumulates class-channel sums; squared-B gives sum-of-squares.
// All class logic is branch-free bitwise math so EXEC stays all-1s.
__global__ __launch_bounds__(32)
void scc_main(const float* __restrict__ color, const float* __restrict__ gt,
              const float* __restrict__ binary, const int* __restrict__ labels,
              float* __restrict__ ws) {
    const int lane = threadIdx.x;
    const int half = lane >> 4;      // K-slot group: 0 -> k{0,1}, 1 -> k{2,3}
    const int r    = lane & 15;      // class row (A) / channel-in-tile (B) / col (D)

    const int n  = blockIdx.x / WPI;
    const int wi = blockIdx.x % WPI;
    const int p0 = wi * CHUNK;

    const float* __restrict__ colN = color  + (size_t)n * CCH * HWPX;
    const float* __restrict__ gtN  = gt     + (size_t)n * HWPX;
    const float* __restrict__ bnN  = binary + (size_t)n * HWPX;
    const int*   __restrict__ lbN  = labels + (size_t)n * HWPX;

    // loop-invariant lane predicates (0/1)
    const unsigned cI = (r < KK) ? 1u : 0u;
    const unsigned i6 = (r == 6) ? 1u : 0u;
    const unsigned i7 = (r == 7) ? 1u : 0u;
    const unsigned i8 = (r == 8) ? 1u : 0u;
    const int      lm = r + 1;

    auto cls = [&](int lab, float g, float b) -> unsigned {
        unsigned s = (g > 0.5f) ? 1u : 0u;
        unsigned t = (b > 0.5f) ? 1u : 0u;
        return (cI & (unsigned)(lab == lm))
             | (i6 & s)
             | (i7 & (s ^ 1u) & (t ^ 1u))
             | (i8 & t & (s ^ 1u));
    };

    v8f acc0 = {}; v8f acc1 = {};    // class x channel sums   (tiles c<16, c>=16)
    v8f asq0 = {}; v8f asq1 = {};    // class x channel sum of squares
    unsigned ucnt = 0;               // exact per-lane count for class r

    for (int q = p0; q < p0 + CHUNK; q += 8) {
        const int p = q + 4 * half;  // this lane's 4 contiguous pixels (16B aligned)

        v4i lab = *(const v4i*)(lbN + p);
        v4f gv  = *(const v4f*)(gtN + p);
        v4f bv  = *(const v4f*)(bnN + p);

        const unsigned a0 = cls(lab.x, gv.x, bv.x);
        const unsigned a1 = cls(lab.y, gv.y, bv.y);
        const unsigned a2 = cls(lab.z, gv.z, bv.z);
        const unsigned a3 = cls(lab.w, gv.w, bv.w);
        ucnt += a0 + a1 + a2 + a3;

        v2f aA = { (float)a0, (float)a1 };   // WMMA step A: pixels p, p+1
        v2f aB = { (float)a2, (float)a3 };   // WMMA step B: pixels p+2, p+3

        v4f c0 = *(const v4f*)(colN + (size_t)r        * HWPX + p);
        v4f c1 = *(const v4f*)(colN + (size_t)(r + 16) * HWPX + p);

        v2f b0A = { c0.x, c0.y }, b0B = { c0.z, c0.w };
        v2f b1A = { c1.x, c1.y }, b1B = { c1.z, c1.w };
        v2f q0A = b0A * b0A, q0B = b0B * b0B;
        v2f q1A = b1A * b1A, q1B = b1B * b1B;

        acc0 = __builtin_amdgcn_wmma_f32_16x16x4_f32(false, aA, false, b0A, (short)0, acc0, false, false);
        acc1 = __builtin_amdgcn_wmma_f32_16x16x4_f32(false, aA, false, b1A, (short)0, acc1, false, false);
        asq0 = __builtin_amdgcn_wmma_f32_16x16x4_f32(false, aA, false, q0A, (short)0, asq0, false, false);
        asq1 = __builtin_amdgcn_wmma_f32_16x16x4_f32(false, aA, false, q1A, (short)0, asq1, false, false);

        acc0 = __builtin_amdgcn_wmma_f32_16x16x4_f32(false, aB, false, b0B, (short)0, acc0, false, false);
        acc1 = __builtin_amdgcn_wmma_f32_16x16x4_f32(false, aB, false, b1B, (short)0, acc1, false, false);
        asq0 = __builtin_amdgcn_wmma_f32_16x16x4_f32(false, aB, false, q0B, (short)0, asq0, false, false);
        asq1 = __builtin_amdgcn_wmma_f32_16x16x4_f32(false, aB, false, q1B, (short)0, asq1, false, false);
    }

    // Flush: D layout row = 8*half + j, col = r. Only class rows 0..8 matter.
    float* S1 = ws + WS_S1  + n * 16 * 32;
    float* SQ = ws + WS_SQ  + n * 16 * 32;
    float* CT = ws + WS_CNT + n * 16;
#pragma unroll
    for (int j = 0; j < 8; ++j) {
        int row = half * 8 + j;
        if (row < 9) {
            atomicAdd(&S1[row * 32 + r],      acc0[j]);
            atomicAdd(&S1[row * 32 + 16 + r], acc1[j]);
            atomicAdd(&SQ[row * 32 + r],      asq0[j]);
            atomicAdd(&SQ[row * 32 + 16 + r], asq1[j]);
        }
    }
    atomicAdd(&CT[r], (float)ucnt);  // lanes r and r+16 together cover all pixels
}

// ---------------- kernel 2: epilogue (means, margins, final scalar) --------
__global__ __launch_bounds__(32)
void scc_final(const float* __restrict__ ws, float* __restrict__ out) {
    __shared__ float part[NB][9];
    const int tid = threadIdx.x;

    if (tid < NB) {
        const int n = tid;
        const float* S1 = ws + WS_S1  + n * 16 * 32;
        const float* SQ = ws + WS_SQ  + n * 16 * 32;
        const float* CT = ws + WS_CNT + n * 16;

        float cnt[KK];
        bool  big[KK];
        int   n_valid = 0;
        for (int k = 0; k < KK; ++k) {
            cnt[k] = CT[k];
            big[k] = cnt[k] >= MINPIX;
            n_valid += big[k] ? 1 : 0;
        }
        const float sub_cnt = CT[6], bg_cnt = CT[7], sc_cnt = CT[8];

        float msub[CCH], mbg[CCH], msc[CCH];
        for (int c = 0; c < CCH; ++c) {
            msub[c] = S1[6 * 32 + c] / (sub_cnt + EPSV);
            mbg[c]  = S1[7 * 32 + c] / (bg_cnt  + EPSV);
            msc[c]  = S1[8 * 32 + c] / (sc_cnt  + EPSV);
        }

        const bool item_valid = (bg_cnt >= MINPIX) && (n_valid >= 1);
        const bool item_multi = item_valid && (n_valid > 1);

        float intra_sum = 0.0f, inter_sum = 0.0f, inst_sum = 0.0f;
        for (int k = 0; k < KK; ++k) {
            if (big[k] && item_valid) {
                const float ic = cnt[k] + EPSV;
                float S2k = 0.0f, dmS = 0.0f, dmm = 0.0f, d2bg = 0.0f, d2sub = 0.0f;
                for (int c = 0; c < CCH; ++c) {
                    const float s = S1[k * 32 + c];
                    const float m = s / ic;
                    S2k += SQ[k * 32 + c];
                    dmS += m * s;
                    dmm += m * m;
                    const float db = m - mbg[c];  d2bg  += db * db;
                    const float ds = m - msub[c]; d2sub += ds * ds;
                }
                intra_sum += (S2k - 2.0f * dmS + cnt[k] * dmm) / ic;
                const float d  = sqrtf(fmaxf(d2bg, 1e-12f));
                const float t  = fmaxf(0.5f - d, 0.0f);
                inter_sum += t * t;
                if (item_multi) inst_sum += d2sub;
            }
        }
        const int inst_n = item_multi ? n_valid : 0;
        const int total  = item_valid ? n_valid : 0;

        float subbg_sum = 0.0f; int subbg_n = 0;
        if (item_valid && sub_cnt >= MINPIX) {
            float d2 = 0.0f;
            for (int c = 0; c < CCH; ++c) { float d = msub[c] - mbg[c]; d2 += d * d; }
            float d = sqrtf(fmaxf(d2, 1e-12f));
            float t = fmaxf(0.5f - d, 0.0f);
            subbg_sum = t * t; subbg_n = 1;
        }
        float subsc_sum = 0.0f; int subsc_n = 0;
        if (item_valid && sc_cnt >= MINPIX && sub_cnt >= MINPIX) {
            float d2 = 0.0f;
            for (int c = 0; c < CCH; ++c) { float d = msub[c] - msc[c]; d2 += d * d; }
            float d = sqrtf(fmaxf(d2, 1e-12f));
            float t = fmaxf(1.0f - d, 0.0f);   // 2*MARGIN
            subsc_sum = t * t; subsc_n = 1;
        }

        part[n][0] = intra_sum;       part[n][1] = inter_sum;
        part[n][2] = inst_sum;        part[n][3] = (float)inst_n;
        part[n][4] = subbg_sum;       part[n][5] = (float)subbg_n;
        part[n][6] = subsc_sum;       part[n][7] = (float)subsc_n;
        part[n][8] = (float)total;
    }
    __syncthreads();

    if (tid == 0) {
        float s_intra = 0, s_inter = 0, s_inst = 0, s_bg = 0, s_sc = 0;
        float n_tot = 0, n_inst = 0, n_bg = 0, n_sc = 0;
        for (int n = 0; n < NB; ++n) {
            s_intra += part[n][0]; s_inter += part[n][1];
            s_inst  += part[n][2]; n_inst  += part[n][3];
            s_bg    += part[n][4]; n_bg    += part[n][5];
            s_sc    += part[n][6]; n_sc    += part[n][7];
            n_tot   += part[n][8];
        }
        auto avg = [](float s, float n) {
            return (n > 0.0f) ? s / fmaxf(n, 1.0f) : 0.0f;
        };
        float loss = avg(s_intra, n_tot)
                   + 0.4f * avg(s_inter, n_tot)
                   + 0.1f * avg(s_inst, n_inst)
                   + 0.2f * avg(s_bg, n_bg)
                   + 0.5f * avg(s_sc, n_sc);
        out[0] = (n_tot > 0.0f) ? loss : 0.0f;
    }
}

// ---------------- host launcher ----------------
extern "C" void kernel_launch(void* const* d_in, const int* in_sizes, int n_in,
                              void* d_out, int out_size, void* d_ws, size_t ws_size,
                              hipStream_t stream) {
    const float* color  = (const float*)d_in[0];
    const float* gt     = (const float*)d_in[1];
    const float* binary = (const float*)d_in[2];
    const int*   labels = (const int*)d_in[3];
    float* ws  = (float*)d_ws;
    float* out = (float*)d_out;

    scc_zero<<<(WS_TOT + 255) / 256, 256, 0, stream>>>(ws);
    scc_main<<<NB * WPI, 32, 0, stream>>>(color, gt, binary, labels, ws);
    scc_final<<<1, 32, 0, stream>>>(ws, out);
}